// WouterSourceGenerator_13434657702539
// MI455X (gfx1250) — compile-verified
//
#include <hip/hip_runtime.h>

// Problem constants (match reference)
#define BB 64
#define NN 10000
#define DD 128
#define FF 4
#define SS 64          // segments of N per batch for the mean reduction
#define ROWS 157       // ceil(NN / SS)
#define EMBW ((FF + 1) * DD)   // 640

typedef __attribute__((ext_vector_type(2))) float v2f;
typedef __attribute__((ext_vector_type(8))) float v8f;

// ---------------------------------------------------------------------------
// Kernel 1: partial sums over N. Block (s, b) sums rows [s*ROWS, min(..,N))
// of batch b; thread d owns column d. Fully coalesced, deterministic order.
// ---------------------------------------------------------------------------
__global__ __launch_bounds__(DD) void seg_sum(const float* __restrict__ H,
                                              float* __restrict__ partial) {
    const int s = blockIdx.x;
    const int b = blockIdx.y;
    const int d = threadIdx.x;

    int n0 = s * ROWS;
    int n1 = n0 + ROWS;
    if (n1 > NN) n1 = NN;
    const int cnt = n1 - n0;

    const float* __restrict__ p = H + ((size_t)b * NN + n0) * DD + d;

    float acc = 0.0f;
    int i = 0;
    for (; i + 4 <= cnt; i += 4) {
        float x0 = p[(size_t)(i + 0) * DD];
        float x1 = p[(size_t)(i + 1) * DD];
        float x2 = p[(size_t)(i + 2) * DD];
        float x3 = p[(size_t)(i + 3) * DD];
        acc += (x0 + x1) + (x2 + x3);
    }
    for (; i < cnt; ++i) acc += p[(size_t)i * DD];

    partial[((size_t)b * SS + s) * DD + d] = acc;
}

// ---------------------------------------------------------------------------
// Kernel 2: combine partials -> mean row of emb; ReLU-gather the F indexed
// rows into emb. emb is (B, 640) row-major.
// ---------------------------------------------------------------------------
__global__ __launch_bounds__(DD) void combine(const float* __restrict__ partial,
                                              const float* __restrict__ H,
                                              const int* __restrict__ indice,
                                              float* __restrict__ emb) {
    const int b = blockIdx.x;
    const int d = threadIdx.x;

    const float* __restrict__ pp = partial + (size_t)b * SS * DD + d;
    float sum = 0.0f;
#pragma unroll 8
    for (int s = 0; s < SS; ++s) sum += pp[(size_t)s * DD];

    float* __restrict__ erow = emb + (size_t)b * EMBW;
    erow[FF * DD + d] = sum * (1.0f / (float)NN);

#pragma unroll
    for (int f = 0; f < FF; ++f) {
        int idx = indice[b * FF + f];
        float v = H[((size_t)b * NN + idx) * DD + d];
        erow[f * DD + d] = fmaxf(v, 0.0f);
    }
}

// ---------------------------------------------------------------------------
// Kernel 3: out = relu(emb @ W) via V_WMMA_F32_16X16X4_F32.
// emb: (64, 640) row-major, W: (640, 128) row-major, out: (64, 128).
// One wave32 per 16x16 output tile; 4 x 8 = 32 tiles; K loop = 640/4 = 160.
//
// Fragment layouts (ISA 7.12.2, wave32):
//   A 16x4 f32 : lane = (K>>1)*16 + M ; VGPR0 = K&~1, VGPR1 = K|1
//   B 4x16 f32 : lane = (K>>1)*16 + N ; VGPR0 = K&~1, VGPR1 = K|1
//   C/D 16x16  : lane = hi*16 + N(lo) ; VGPR v holds M = v + 8*hi
// ---------------------------------------------------------------------------
__global__ __launch_bounds__(32) void gemm_relu(const float* __restrict__ emb,
                                                const float* __restrict__ W,
                                                float* __restrict__ out) {
    const int tile = blockIdx.x;      // 0..31
    const int mt   = tile >> 3;       // 4 M-tiles
    const int nt   = tile & 7;        // 8 N-tiles
    const int lane = threadIdx.x;     // 0..31
    const int hi   = lane >> 4;       // 0 | 1  -> K pair select
    const int lo   = lane & 15;       // M (for A) / N (for B,C,D)

    const int m = mt * 16 + lo;
    const int n = nt * 16 + lo;

    const float* __restrict__ arow = emb + (size_t)m * EMBW;

    v8f c = {};
#pragma unroll 4
    for (int k0 = 0; k0 < EMBW; k0 += 4) {
        const int ka = k0 + hi * 2;
        v2f a, bfrag;
        a.x     = arow[ka];
        a.y     = arow[ka + 1];
        bfrag.x = W[(size_t)ka * DD + n];
        bfrag.y = W[(size_t)(ka + 1) * DD + n];
        // 8 args: (neg_a, A, neg_b, B, c_mod, C, reuse_a, reuse_b)
        c = __builtin_amdgcn_wmma_f32_16x16x4_f32(
                false, a, false, bfrag, (short)0, c, false, false);
    }

#pragma unroll
    for (int v = 0; v < 8; ++v) {
        const int row = mt * 16 + v + hi * 8;
        out[(size_t)row * DD + n] = fmaxf(c[v], 0.0f);
    }
}

// ---------------------------------------------------------------------------
// Launch
// ---------------------------------------------------------------------------
extern "C" void kernel_launch(void* const* d_in, const int* in_sizes, int n_in,
                              void* d_out, int out_size, void* d_ws, size_t ws_size,
                              hipStream_t stream) {
    const float* H      = (const float*)d_in[0];   // (B, N, D) fp32
    const int*   indice = (const int*)d_in[1];     // (B, F) int
    const float* W      = (const float*)d_in[2];   // (640, 128) fp32
    float*       out    = (float*)d_out;           // (B, 1, D) fp32

    float* partial = (float*)d_ws;                         // B*S*D floats
    float* emb     = partial + (size_t)BB * SS * DD;       // B*640 floats

    dim3 g1(SS, BB);
    seg_sum<<<g1, DD, 0, stream>>>(H, partial);
    combine<<<BB, DD, 0, stream>>>(partial, H, indice, emb);
    gemm_relu<<<32, 32, 0, stream>>>(emb, W, out);
}